// MambaBlock_46540265620202
// MI455X (gfx1250) — compile-verified
//
#include <hip/hip_runtime.h>
#include <math.h>

// ---------------- problem constants ----------------
#define BB      2
#define LL      1024
#define DMODEL  1024
#define DSTATE  64
#define DCONV   4
#define HEADD   64
#define DINNER  2048
#define NHEADS  32
#define CONVDIM 2176          // DINNER + 2*DSTATE
#define DINPROJ 4256          // 2*DINNER + 2*DSTATE + NHEADS
#define MTOK    (BB*LL)       // 2048 token rows
#define CT      64            // chunk length (SSD)
#define NC      (LL/CT)       // 16 chunks per batch
#define LDSP    66            // padded LDS row stride (even: keeps b64 align, kills conflicts)

typedef __attribute__((ext_vector_type(2))) float v2f;
typedef __attribute__((ext_vector_type(8))) float v8f;

__device__ __forceinline__ float sigf(float x) { return 1.0f / (1.0f + expf(-x)); }

#define WMMA_F32(a, b, c) \
    __builtin_amdgcn_wmma_f32_16x16x4_f32(false, (a), false, (b), (short)0, (c), false, false)

// ---------------------------------------------------------------------------
// Dense WMMA GEMM:  C[M,N] = A[M,K] @ W[N,K]^T   (fp32, V_WMMA_F32_16X16X4_F32)
// one wave per block; TMxTN register tiling (tiles of 16).
// grid = (N/(TN*16), M/(TM*16)), block = 32
// ---------------------------------------------------------------------------
template <int TM, int TN>
__global__ void wmma_gemm_xwt(const float* __restrict__ A,
                              const float* __restrict__ W,
                              float* __restrict__ C,
                              int M, int N, int K)
{
    const int lane = threadIdx.x & 31;
    const int lo   = lane & 15;
    const int hi   = lane >> 4;
    const int tn   = blockIdx.x * (TN * 16);
    const int tm   = blockIdx.y * (TM * 16);

    const float* ar[TM];
    const float* wr[TN];
#pragma unroll
    for (int i = 0; i < TM; ++i) ar[i] = A + (size_t)(tm + i * 16 + lo) * K;
#pragma unroll
    for (int j = 0; j < TN; ++j) wr[j] = W + (size_t)(tn + j * 16 + lo) * K;

    v8f acc[TM][TN];
#pragma unroll
    for (int i = 0; i < TM; ++i)
#pragma unroll
        for (int j = 0; j < TN; ++j) acc[i][j] = (v8f){};

    for (int k = 0; k < K; k += 4) {
        v2f fa[TM], fb[TN];
#pragma unroll
        for (int i = 0; i < TM; ++i) fa[i] = *(const v2f*)(ar[i] + k + 2 * hi);
#pragma unroll
        for (int j = 0; j < TN; ++j) fb[j] = *(const v2f*)(wr[j] + k + 2 * hi);
#pragma unroll
        for (int i = 0; i < TM; ++i)
#pragma unroll
            for (int j = 0; j < TN; ++j)
                acc[i][j] = WMMA_F32(fa[i], fb[j], acc[i][j]);
    }
#pragma unroll
    for (int i = 0; i < TM; ++i)
#pragma unroll
        for (int r = 0; r < 8; ++r) {
            const size_t ro = (size_t)(tm + i * 16 + r + 8 * hi) * N;
#pragma unroll
            for (int j = 0; j < TN; ++j)
                C[ro + tn + j * 16 + lo] = acc[i][j][r];
        }
}

// ---------------------------------------------------------------------------
// RMSNorm over last dim D (D <= 2048), one block (256 thr) per row
// ---------------------------------------------------------------------------
__global__ void rmsnorm_kernel(const float* __restrict__ in,
                               const float* __restrict__ w,
                               float* __restrict__ out, int D, float eps)
{
    __shared__ float red[256];
    const size_t row = blockIdx.x;
    const float* x = in + row * (size_t)D;
    float vals[8];
    float ss = 0.f;
    int j = 0;
    for (int i = threadIdx.x; i < D; i += 256, ++j) { float v = x[i]; vals[j] = v; ss += v * v; }
    red[threadIdx.x] = ss; __syncthreads();
    for (int s = 128; s > 0; s >>= 1) {
        if ((int)threadIdx.x < s) red[threadIdx.x] += red[threadIdx.x + s];
        __syncthreads();
    }
    const float scale = rsqrtf(red[0] / (float)D + eps);
    j = 0;
    for (int i = threadIdx.x; i < D; i += 256, ++j)
        out[row * (size_t)D + i] = vals[j] * scale * w[i];
}

// ---------------------------------------------------------------------------
// gated norm: y <- rmsnorm(y * silu(z), w, 1e-5), z = zx[row*DINPROJ + d]
// ---------------------------------------------------------------------------
__global__ void gated_rmsnorm_kernel(float* __restrict__ y,
                                     const float* __restrict__ zx,
                                     const float* __restrict__ w)
{
    __shared__ float red[256];
    const size_t row = blockIdx.x;
    float* yr = y + row * (size_t)DINNER;
    const float* zr = zx + row * (size_t)DINPROJ;
    float vals[8];
    float ss = 0.f;
    int j = 0;
    for (int i = threadIdx.x; i < DINNER; i += 256, ++j) {
        const float z = zr[i];
        const float v = yr[i] * (z * sigf(z));
        vals[j] = v; ss += v * v;
    }
    red[threadIdx.x] = ss; __syncthreads();
    for (int s = 128; s > 0; s >>= 1) {
        if ((int)threadIdx.x < s) red[threadIdx.x] += red[threadIdx.x + s];
        __syncthreads();
    }
    const float scale = rsqrtf(red[0] / (float)DINNER + 1e-5f);
    j = 0;
    for (int i = threadIdx.x; i < DINNER; i += 256, ++j)
        yr[i] = vals[j] * scale * w[i];
}

// ---------------------------------------------------------------------------
// causal depthwise conv (4 taps) + SiLU over the xBC slice of zxbcdt
// ---------------------------------------------------------------------------
__global__ void conv_silu_kernel(const float* __restrict__ zx,
                                 const float* __restrict__ cw,
                                 const float* __restrict__ cb,
                                 float* __restrict__ xbc)
{
    const size_t idx = (size_t)blockIdx.x * 256 + threadIdx.x;
    if (idx >= (size_t)MTOK * CONVDIM) return;
    const int    c = (int)(idx % CONVDIM);
    const size_t m = idx / CONVDIM;
    const int    l = (int)(m & (LL - 1));
    float acc = cb[c];
#pragma unroll
    for (int i = 0; i < DCONV; ++i) {
        const int ll = l - (DCONV - 1) + i;
        if (ll >= 0)
            acc += zx[(m - (size_t)(DCONV - 1 - i)) * DINPROJ + DINNER + c] * cw[c * DCONV + i];
    }
    xbc[idx] = acc * sigf(acc);
}

// ---------------------------------------------------------------------------
// dt = softplus(dt_raw + dt_bias)
// ---------------------------------------------------------------------------
__global__ void dt_kernel(const float* __restrict__ zx,
                          const float* __restrict__ dt_bias,
                          float* __restrict__ dtO)
{
    const int idx = blockIdx.x * 256 + threadIdx.x;          // over MTOK*NHEADS
    if (idx >= MTOK * NHEADS) return;
    const int    h = idx & (NHEADS - 1);
    const size_t m = (size_t)(idx >> 5);
    const float  x = zx[m * DINPROJ + DINNER + CONVDIM + h] + dt_bias[h];
    dtO[idx] = (x > 20.f) ? x : log1pf(expf(x));
}

// ---------------------------------------------------------------------------
// per-chunk inclusive cumsum of dtA = dt * (-exp(A_log))
// grid = B*H*NC (idx = (b*NHEADS+h)*NC + c), block = 64
// ---------------------------------------------------------------------------
__global__ void cum_kernel(const float* __restrict__ dtb,
                           const float* __restrict__ A_log,
                           float* __restrict__ cum, float* __restrict__ tot)
{
    const int idx = blockIdx.x;
    const int c   = idx & (NC - 1);
    const int bh  = idx >> 4;
    const int h   = bh & (NHEADS - 1);
    const int b   = bh >> 5;
    const int t   = threadIdx.x;
    const size_t m = (size_t)b * LL + (size_t)c * CT + t;
    __shared__ float s[CT];
    s[t] = dtb[m * NHEADS + h] * (-expf(A_log[h]));
    __syncthreads();
    for (int off = 1; off < CT; off <<= 1) {
        const float add = (t >= off) ? s[t - off] : 0.f;
        __syncthreads();
        s[t] += add;
        __syncthreads();
    }
    cum[(size_t)idx * CT + t] = s[t];
    if (t == CT - 1) tot[idx] = s[t];
}

// ---------------------------------------------------------------------------
// G[b,c][t][tau] = sum_n C[t][n] * B[tau][n]   (head-independent, NGROUPS==1)
// grid = B*NC, block = 128 (4 waves; wave -> t-tile, loop tau-tiles)
// ---------------------------------------------------------------------------
__global__ void chunk_G_kernel(const float* __restrict__ xbc, float* __restrict__ G)
{
    const int bc   = blockIdx.x;
    const int b    = bc >> 4, c = bc & (NC - 1);
    const int wave = threadIdx.x >> 5;
    const int lane = threadIdx.x & 31;
    const int lo   = lane & 15, hi = lane >> 4;

    const size_t base = (size_t)b * LL + (size_t)c * CT;
    const float* arow = xbc + (base + wave * 16 + lo) * CONVDIM + DINNER + DSTATE; // C row
    const float* br[4];
#pragma unroll
    for (int j = 0; j < 4; ++j)
        br[j] = xbc + (base + j * 16 + lo) * CONVDIM + DINNER;                     // B row

    v8f acc[4] = {};
    for (int k = 0; k < DSTATE; k += 4) {
        const v2f a = *(const v2f*)(arow + k + 2 * hi);
#pragma unroll
        for (int j = 0; j < 4; ++j) {
            const v2f bb = *(const v2f*)(br[j] + k + 2 * hi);
            acc[j] = WMMA_F32(a, bb, acc[j]);
        }
    }
    float* Go = G + (size_t)bc * (CT * CT);
#pragma unroll
    for (int j = 0; j < 4; ++j)
#pragma unroll
        for (int r = 0; r < 8; ++r)
            Go[(wave * 16 + r + 8 * hi) * CT + j * 16 + lo] = acc[j][r];
}

// ---------------------------------------------------------------------------
// chunk state: S[n][p] = sum_tau exp(tot - cum[tau]) * dt[tau] * B[tau][n] * X[tau][p]
// grid = B*NC*H (idx = (b*NC+c)*NHEADS + h), block = 128
// staging: coalesced global reads, transposed LDS writes (stride LDSP)
// ---------------------------------------------------------------------------
__global__ void chunk_state_kernel(const float* __restrict__ xbc,
                                   const float* __restrict__ dtb,
                                   const float* __restrict__ cum,
                                   const float* __restrict__ tot,
                                   float* __restrict__ S)
{
    const int idx = blockIdx.x;
    const int h   = idx & (NHEADS - 1);
    const int bc  = idx >> 5;
    const int c   = bc & (NC - 1);
    const int b   = bc >> 4;
    const size_t base   = (size_t)b * LL + (size_t)c * CT;
    const int    cumidx = (b * NHEADS + h) * NC + c;

    __shared__ float Alds[CT * LDSP];   // [n][tau] = B[tau][n] * w[tau]
    __shared__ float Xt[CT * LDSP];     // [p][tau] = X[tau][p]
    __shared__ float wv[CT];

    if (threadIdx.x < CT) {
        const int tau = threadIdx.x;
        wv[tau] = expf(tot[cumidx] - cum[(size_t)cumidx * CT + tau]) *
                  dtb[(base + tau) * NHEADS + h];
    }
    __syncthreads();
    for (int e = threadIdx.x; e < CT * CT; e += 128) {
        const int tau = e >> 6, r = e & 63;                 // r consecutive per lane
        const float* row = xbc + (base + tau) * CONVDIM;
        Alds[r * LDSP + tau] = row[DINNER + r] * wv[tau];   // coalesced read, transposed write
        Xt[r * LDSP + tau]   = row[h * HEADD + r];
    }
    __syncthreads();

    const int wave = threadIdx.x >> 5;
    const int lane = threadIdx.x & 31;
    const int lo   = lane & 15, hi = lane >> 4;

    const float* arow = Alds + (wave * 16 + lo) * LDSP;
    v8f acc[4] = {};
    for (int k = 0; k < CT; k += 4) {
        const v2f a = *(const v2f*)(arow + k + 2 * hi);
#pragma unroll
        for (int j = 0; j < 4; ++j) {
            const v2f bb = *(const v2f*)(Xt + (j * 16 + lo) * LDSP + k + 2 * hi);
            acc[j] = WMMA_F32(a, bb, acc[j]);
        }
    }
    float* So = S + (size_t)idx * (DSTATE * HEADD);
#pragma unroll
    for (int j = 0; j < 4; ++j)
#pragma unroll
        for (int r = 0; r < 8; ++r)
            So[(wave * 16 + r + 8 * hi) * HEADD + j * 16 + lo] = acc[j][r];
}

// ---------------------------------------------------------------------------
// chunk-state recurrence (16 serial steps): Hst[c] = state BEFORE chunk c
// grid = B*H (bh = b*NHEADS+h), block = 256
// ---------------------------------------------------------------------------
__global__ void hstate_kernel(const float* __restrict__ S,
                              const float* __restrict__ tot,
                              float* __restrict__ Hst)
{
    const int bh = blockIdx.x;
    const int b  = bh >> 5, h = bh & (NHEADS - 1);
    float H[16];
#pragma unroll
    for (int i = 0; i < 16; ++i) H[i] = 0.f;
    for (int c = 0; c < NC; ++c) {
        const size_t off = ((size_t)(b * NC + c) * NHEADS + h) * (DSTATE * HEADD);
        const float  e   = expf(tot[bh * NC + c]);
#pragma unroll
        for (int i = 0; i < 16; ++i) {
            const int e2 = threadIdx.x + i * 256;
            Hst[off + e2] = H[i];
            H[i] = H[i] * e + S[off + e2];
        }
    }
}

// ---------------------------------------------------------------------------
// chunk output: Y = (G.decay.dt masked) @ X  +  exp(cum)*C @ Hst  +  D*X
// grid = B*NC*H (idx = (b*NC+c)*NHEADS + h), block = 128
// ---------------------------------------------------------------------------
__global__ void chunk_Y_kernel(const float* __restrict__ xbc,
                               const float* __restrict__ dtb,
                               const float* __restrict__ cum,
                               const float* __restrict__ G,
                               const float* __restrict__ Hst,
                               const float* __restrict__ Dp,
                               float* __restrict__ y)
{
    const int idx = blockIdx.x;
    const int h   = idx & (NHEADS - 1);
    const int bc  = idx >> 5;
    const int c   = bc & (NC - 1);
    const int b   = bc >> 4;
    const size_t base   = (size_t)b * LL + (size_t)c * CT;
    const int    cumidx = (b * NHEADS + h) * NC + c;

    __shared__ float Albuf[CT * LDSP];  // phase1: masked M [t][tau]; phase2: Cw [t][n]
    __shared__ float Xt[CT * LDSP];     // [p][t] = X[t][p]
    __shared__ float Hb[CT * LDSP];     // [p][n] = Hst[n][p]
    __shared__ float cumv[CT], dtv[CT];

    if (threadIdx.x < CT) {
        const int t = threadIdx.x;
        cumv[t] = cum[(size_t)cumidx * CT + t];
        dtv[t]  = dtb[(base + t) * NHEADS + h];
    }
    __syncthreads();

    const float* Gi = G + (size_t)bc * (CT * CT);
    for (int e = threadIdx.x; e < CT * CT; e += 128) {
        const int rr = e >> 6, q = e & 63;                  // q consecutive per lane
        // M[t=rr][tau=q] (row-major write, coalesced G read)
        Albuf[rr * LDSP + q] = (q <= rr) ? Gi[e] * expf(cumv[rr] - cumv[q]) * dtv[q] : 0.f;
        // Xt[p=q][t=rr] (coalesced xbc read, transposed write)
        Xt[q * LDSP + rr] = xbc[(base + rr) * CONVDIM + h * HEADD + q];
    }
    __syncthreads();

    const int wave = threadIdx.x >> 5;
    const int lane = threadIdx.x & 31;
    const int lo   = lane & 15, hi = lane >> 4;
    const int tm   = wave * 16;

    v8f acc[4] = {};
    {   // phase 1: Y_intra = M @ X
        const float* arow = Albuf + (tm + lo) * LDSP;
        for (int k = 0; k < CT; k += 4) {
            const v2f a = *(const v2f*)(arow + k + 2 * hi);
#pragma unroll
            for (int j = 0; j < 4; ++j) {
                const v2f bb = *(const v2f*)(Xt + (j * 16 + lo) * LDSP + k + 2 * hi);
                acc[j] = WMMA_F32(a, bb, acc[j]);
            }
        }
    }
    __syncthreads();
    {   // rebuild LDS: Albuf <- Cw[t][n] = C[t][n]*exp(cum[t]); Hb <- Hst^T
        const float* Ho = Hst + (size_t)idx * (DSTATE * HEADD);
        for (int e = threadIdx.x; e < CT * CT; e += 128) {
            const int rr = e >> 6, q = e & 63;
            Albuf[rr * LDSP + q] = xbc[(base + rr) * CONVDIM + DINNER + DSTATE + q] * expf(cumv[rr]);
            Hb[q * LDSP + rr]    = Ho[rr * HEADD + q];      // [p=q][n=rr]
        }
    }
    __syncthreads();
    {   // phase 2: Y += Cw @ Hst
        const float* arow = Albuf + (tm + lo) * LDSP;
        for (int k = 0; k < DSTATE; k += 4) {
            const v2f a = *(const v2f*)(arow + k + 2 * hi);
#pragma unroll
            for (int j = 0; j < 4; ++j) {
                const v2f bb = *(const v2f*)(Hb + (j * 16 + lo) * LDSP + k + 2 * hi);
                acc[j] = WMMA_F32(a, bb, acc[j]);
            }
        }
    }
    const float Dh = Dp[h];
#pragma unroll
    for (int j = 0; j < 4; ++j)
#pragma unroll
        for (int r = 0; r < 8; ++r) {
            const int t = tm + r + 8 * hi;
            const int p = j * 16 + lo;
            y[(base + t) * DINNER + h * HEADD + p] = acc[j][r] + Dh * Xt[p * LDSP + t];
        }
}

// ---------------------------------------------------------------------------
// elementwise helpers
// ---------------------------------------------------------------------------
__global__ void add_kernel(const float* __restrict__ a, const float* __restrict__ b,
                           float* __restrict__ o, int n)
{
    const int i = blockIdx.x * 256 + threadIdx.x;
    if (i < n) o[i] = a[i] + b[i];
}

__global__ void final_kernel(const float* __restrict__ g, const float* __restrict__ h2,
                             const float* __restrict__ xa, const float* __restrict__ xf,
                             const float* __restrict__ xd,
                             float* __restrict__ out0, float* __restrict__ out1, int n)
{
    const int i = blockIdx.x * 256 + threadIdx.x;
    if (i < n) {
        const float gv = g[i];
        const float f  = gv * sigf(gv) * h2[i];
        out0[i] = xa[i] + f;
        out1[i] = xd[i] + xf[i] + f;
    }
}

// ---------------------------------------------------------------------------
extern "C" void kernel_launch(void* const* d_in, const int* in_sizes, int n_in,
                              void* d_out, int out_size, void* d_ws, size_t ws_size,
                              hipStream_t stream)
{
    const float* x       = (const float*)d_in[0];
    const float* x_d     = (const float*)d_in[1];
    const float* norm_w  = (const float*)d_in[2];
    const float* W_in    = (const float*)d_in[3];
    const float* conv_w  = (const float*)d_in[4];
    const float* conv_b  = (const float*)d_in[5];
    const float* dt_bias = (const float*)d_in[6];
    const float* A_log   = (const float*)d_in[7];
    const float* D_param = (const float*)d_in[8];
    const float* ssm_nw  = (const float*)d_in[9];
    const float* W_out   = (const float*)d_in[10];
    const float* w1      = (const float*)d_in[11];
    const float* w2      = (const float*)d_in[12];

    float* out0 = (float*)d_out;
    float* out1 = out0 + (size_t)MTOK * DMODEL;

    // workspace layout (floats)
    float* ws   = (float*)d_ws;
    float* xln  = ws;                                    // 2,097,152 (reused for x_ln2)
    float* zx   = xln  + (size_t)MTOK * DMODEL;          // 8,716,288 (reused for g/h2)
    float* xbc  = zx   + (size_t)MTOK * DINPROJ;         // 4,456,448
    float* dtb  = xbc  + (size_t)MTOK * CONVDIM;         //    65,536
    float* cumb = dtb  + (size_t)MTOK * NHEADS;          //    65,536
    float* totb = cumb + (size_t)MTOK * NHEADS;          //     1,024
    float* Gb   = totb + (size_t)BB * NHEADS * NC;       //   131,072
    float* Sb   = Gb   + (size_t)BB * NC * CT * CT;      // 4,194,304
    float* Hstb = Sb   + (size_t)BB * NC * NHEADS * DSTATE * HEADD;  // 4,194,304
    float* yb   = Hstb + (size_t)BB * NC * NHEADS * DSTATE * HEADD;  // 4,194,304
    float* xf   = yb   + (size_t)MTOK * DINNER;          // 2,097,152
    float* xa   = xf   + (size_t)MTOK * DMODEL;          // 2,097,152
    float* gb   = zx;                                    // reuse zxbcdt space
    float* hb   = zx + (size_t)MTOK * DMODEL;

    const int nmod = MTOK * DMODEL;

    // 1) x_ln = rmsnorm(x)
    rmsnorm_kernel<<<MTOK, 256, 0, stream>>>(x, norm_w, xln, DMODEL, 1e-6f);

    // 2) zxbcdt = x_ln @ W_in^T    (64Mx32N per wave)
    wmma_gemm_xwt<4, 2><<<dim3(DINPROJ / 32, MTOK / 64), 32, 0, stream>>>(
        xln, W_in, zx, MTOK, DINPROJ, DMODEL);

    // 3) causal depthwise conv + silu
    conv_silu_kernel<<<((size_t)MTOK * CONVDIM + 255) / 256, 256, 0, stream>>>(
        zx, conv_w, conv_b, xbc);

    // 4) dt softplus
    dt_kernel<<<(MTOK * NHEADS + 255) / 256, 256, 0, stream>>>(zx, dt_bias, dtb);

    // 5) per-chunk cumsum of dtA
    cum_kernel<<<BB * NHEADS * NC, CT, 0, stream>>>(dtb, A_log, cumb, totb);

    // 6) G = C @ B^T per (b,c)          [WMMA]
    chunk_G_kernel<<<BB * NC, 128, 0, stream>>>(xbc, Gb);

    // 7) chunk states S per (b,c,h)     [WMMA]
    chunk_state_kernel<<<BB * NC * NHEADS, 128, 0, stream>>>(xbc, dtb, cumb, totb, Sb);

    // 8) 16-step chunk-state recurrence
    hstate_kernel<<<BB * NHEADS, 256, 0, stream>>>(Sb, totb, Hstb);

    // 9) Y = intra + inter + D*X        [WMMA]
    chunk_Y_kernel<<<BB * NC * NHEADS, 128, 0, stream>>>(
        xbc, dtb, cumb, Gb, Hstb, D_param, yb);

    // 10) y = rmsnorm(y * silu(z))
    gated_rmsnorm_kernel<<<MTOK, 256, 0, stream>>>(yb, zx, ssm_nw);

    // 11) x_f = y @ W_out^T
    wmma_gemm_xwt<4, 2><<<dim3(DMODEL / 32, MTOK / 64), 32, 0, stream>>>(
        yb, W_out, xf, MTOK, DMODEL, DINNER);

    // 12) x_a = x + x_f
    add_kernel<<<(nmod + 255) / 256, 256, 0, stream>>>(x, xf, xa, nmod);

    // 13) x_ln2 = rmsnorm(x_a)
    rmsnorm_kernel<<<MTOK, 256, 0, stream>>>(xa, norm_w, xln, DMODEL, 1e-6f);

    // 14) MLP GEMMs
    wmma_gemm_xwt<4, 2><<<dim3(DMODEL / 32, MTOK / 64), 32, 0, stream>>>(
        xln, w1, gb, MTOK, DMODEL, DMODEL);
    wmma_gemm_xwt<4, 2><<<dim3(DMODEL / 32, MTOK / 64), 32, 0, stream>>>(
        xln, w2, hb, MTOK, DMODEL, DMODEL);

    // 15) outputs
    final_kernel<<<(nmod + 255) / 256, 256, 0, stream>>>(
        gb, hb, xa, xf, x_d, out0, out1, nmod);
}